// DotProductPredictor_13804024889621
// MI455X (gfx1250) — compile-verified
//
#include <hip/hip_runtime.h>

// MI455X / gfx1250, wave32.
// Edge scoring: out[e] = sigmoid( dot(features[src[e]], features[dst[e]]) ), D=64.
//
// Formulation: score = (U .* V) x ones  via V_WMMA_F32_16X16X4_F32 (full f32),
// B = all-ones so no transpose of the gathered data is ever needed, and the
// f32 16x4 A-operand layout matches a plain b64 gather per lane.

typedef __attribute__((ext_vector_type(2))) float v2f;
typedef __attribute__((ext_vector_type(8))) float v8f;

#define D_FEAT 64
#define TILE   16   // edges per wave

__global__ __launch_bounds__(256) void edge_dot_sigmoid_wmma(
    const float* __restrict__ feat,   // [N_NODES, 64] f32
    const int*   __restrict__ src,    // [E] i32
    const int*   __restrict__ dst,    // [E] i32
    float*       __restrict__ out,    // [E] f32
    int n_edges)
{
    const int lane         = threadIdx.x & 31;
    const int waveInBlock  = threadIdx.x >> 5;
    const int wavesPerBlk  = blockDim.x >> 5;
    const int tile         = blockIdx.x * wavesPerBlk + waveInBlock;   // wave-uniform
    const int nTiles       = (n_edges + TILE - 1) / TILE;
    if (tile >= nTiles) return;                                        // whole-wave exit

    const int m    = lane & 15;   // edge-in-tile (rows duplicated across wave halves)
    const int half = lane >> 4;   // 0: K=0,1 of each chunk; 1: K=2,3 (f32 A layout)

    const int e        = tile * TILE + m;
    const int eClamped = (e < n_edges) ? e : (n_edges - 1);

    // Row bases; lanes 0-15 and 16-31 read the same 16 rows at offsets 0 / +8B.
    const float* uRow = feat + (long long)src[eClamped] * D_FEAT + half * 2;
    const float* vRow = feat + (long long)dst[eClamped] * D_FEAT + half * 2;

    v8f acc = {};                       // 16x16 f32 C/D: VGPR r = score[r] (lanes 0-15),
                                        //                         score[r+8] (lanes 16-31)
    v2f ones; ones[0] = 1.0f; ones[1] = 1.0f;   // B = ones(4x16): layout-independent

    // 16 K-chunks of 4 cover D=64. Each chunk: b64 gather of U and V slices in
    // native A-operand layout, elementwise multiply, WMMA row-sum into acc.
#pragma unroll
    for (int c = 0; c < 16; ++c) {
        v2f u = *(const v2f*)(uRow + 4 * c);
        v2f v = *(const v2f*)(vRow + 4 * c);
        v2f w = u * v;
        // 8 args: (neg_a, A, neg_b, B, c_mod, C, reuse_a, reuse_b)
        acc = __builtin_amdgcn_wmma_f32_16x16x4_f32(
            false, w, false, ones, (short)0, acc, false, false);
    }

    // Compact: lane l should end with acc[l&7] (score[l&7] in low half,
    // score[(l&7)+8] in high half) -> 3-level v_cndmask tree.
    const bool b0 = (lane & 1) != 0;
    const bool b1 = (lane & 2) != 0;
    const bool b2 = (lane & 4) != 0;
    float t0 = b0 ? acc[1] : acc[0];
    float t1 = b0 ? acc[3] : acc[2];
    float t2 = b0 ? acc[5] : acc[4];
    float t3 = b0 ? acc[7] : acc[6];
    float s0 = b1 ? t1 : t0;
    float s1 = b1 ? t3 : t2;
    float x  = b2 ? s1 : s0;
    // Lanes 0-7 already hold score[l]; lanes 8-15 fetch score[l] from lane l+8
    // (high half holds scores 8-15). One ds_bpermute via __shfl.
    const int srcLane = (lane & 8) ? (lane + 8) : lane;
    float score = __shfl(x, srcLane, 32);

    // Sigmoid (f32). exp overflow -> inf -> sigmoid -> 0: correct limit.
    float sig = 1.0f / (1.0f + __expf(-score));

    if (half == 0 && e < n_edges)
        out[e] = sig;   // lanes 0-15: 64B half-coalesced store
}

extern "C" void kernel_launch(void* const* d_in, const int* in_sizes, int n_in,
                              void* d_out, int out_size, void* d_ws, size_t ws_size,
                              hipStream_t stream) {
    const float* feat = (const float*)d_in[0];
    const int*   src  = (const int*)d_in[1];
    const int*   dst  = (const int*)d_in[2];
    float*       out  = (float*)d_out;

    const int n_edges = in_sizes[1];
    const int nTiles  = (n_edges + TILE - 1) / TILE;
    const int threads = 256;                       // 8 waves -> 8 tiles per block
    const int wavesPerBlk = threads / 32;
    const int blocks  = (nTiles + wavesPerBlk - 1) / wavesPerBlk;

    edge_dot_sigmoid_wmma<<<blocks, threads, 0, stream>>>(feat, src, dst, out, n_edges);
}